// TFNLayer_34033320853621
// MI455X (gfx1250) — compile-verified
//
#include <hip/hip_runtime.h>
#include <hip/hip_bf16.h>

// ---------------------------------------------------------------------------
// TFN layer for MI455X (gfx1250, wave32).
//  * All GEMMs on V_WMMA_F32_16X16X4_F32 (f32-faithful; problem is
//    bandwidth/atomic bound, ~4 GFLOP vs ~0.5 GB @ 23.3 TB/s, so low-precision
//    WMMA buys nothing).
//  * Weight staging into LDS uses GLOBAL_LOAD_ASYNC_TO_LDS_B128 (ASYNCcnt)
//    per CDNA5 ISA — no VGPR bounce.
//  * segment_sum via f32 global atomics into a 20.5 MB L2-resident buffer.
// ---------------------------------------------------------------------------

#define NN 10000
#define NE 320000

typedef __attribute__((ext_vector_type(2))) float v2f;
typedef __attribute__((ext_vector_type(8))) float v8f;

__device__ __forceinline__ v8f wmma4(v2f a, v2f b, v8f c) {
#if __has_builtin(__builtin_amdgcn_wmma_f32_16x16x4_f32)
  return __builtin_amdgcn_wmma_f32_16x16x4_f32(false, a, false, b, (short)0, c,
                                               false, false);
#else
  c[0] += a.x * b.x + a.y * b.y;  // unreachable on gfx1250 (probe-confirmed)
  return c;
#endif
}

// Async global->LDS copy, 16B per lane. vdst = LDS byte address (low 32 bits
// of a generic pointer to __shared__), vaddr = 64-bit global address.
__device__ __forceinline__ void async_copy_b128(unsigned lds_addr,
                                                const void* gaddr) {
  asm volatile("global_load_async_to_lds_b128 %0, %1, off"
               :
               : "v"(lds_addr), "v"(gaddr)
               : "memory");
}

__device__ __forceinline__ void wait_async0() {
#if __has_builtin(__builtin_amdgcn_s_wait_asynccnt)
  __builtin_amdgcn_s_wait_asynccnt(0);
#else
  asm volatile("s_wait_asynccnt 0x0" ::: "memory");
#endif
}

__device__ __forceinline__ unsigned lds_off(const void* p) {
  return (unsigned)(unsigned long long)p;  // low 32 bits = LDS byte address
}

// shifted softplus: log(1+exp(x)) - log(2), numerically stable
__device__ __forceinline__ float sspf(float x) {
  float sp = fmaxf(x, 0.0f) + log1pf(__expf(-fabsf(x)));
  return sp - 0.6931471805599453f;
}

#define INV_SQRT8   0.35355339059327373f
#define INV_SQRT32  0.17677669529663687f
#define INV_SQRT3   0.57735026918962576f
#define INV_SQRT128 0.08838834764831845f
#define INV_SQRT256 0.0625f

// ===========================================================================
// Kernel 1: x0 = f0 @ W1_0 / 8 ; x1[:,:,c] = f1[:,:,c] @ W1_1 / 8
// Block: 128 thr (4 waves), 16 nodes. wave0 -> x0, waves1..3 -> x1 c=0..2
// Weights staged with async-to-LDS; A blocks staged through VGPRs (the f1
// block needs a [n][u][c] -> [c][m][u] transpose).
// ===========================================================================
__global__ __launch_bounds__(128) void k_node_linear(
    const float* __restrict__ f0, const float* __restrict__ f1,
    const float* __restrict__ W10, const float* __restrict__ W11,
    float* __restrict__ x0, float* __restrict__ x1) {
  __shared__ float sA0[16 * 64];
  __shared__ float sA1[3 * 16 * 64];
  __shared__ float sW0[64 * 64];
  __shared__ float sW1[64 * 64];

  const int tid = threadIdx.x;
  const int nbase = blockIdx.x * 16;

  // --- async-stage both 64x64 weight matrices (1024 b128 transfers each) ---
  {
    const unsigned l0 = lds_off(sW0), l1 = lds_off(sW1);
    const char* g0 = (const char*)W10;
    const char* g1 = (const char*)W11;
    for (int i = tid; i < 1024; i += 128) {
      async_copy_b128(l0 + i * 16, g0 + i * 16);
      async_copy_b128(l1 + i * 16, g1 + i * 16);
    }
  }
  // --- stage A0 block (16x64) as float4 ---
  {
    const float4* f0v = (const float4*)(f0 + (size_t)nbase * 64);
    float4* a0v = (float4*)sA0;
    for (int i = tid; i < 256; i += 128) a0v[i] = f0v[i];
  }
  // --- stage A1 as [c][m][u] (transpose of [n][u][c]) ---
  for (int i = tid; i < 16 * 64 * 3; i += 128) {
    int m = i / 192, rem = i % 192, u = rem / 3, c = rem % 3;
    sA1[(c * 16 + m) * 64 + u] = f1[(size_t)(nbase + m) * 192 + u * 3 + c];
  }
  wait_async0();
  __syncthreads();

  const int wave = tid >> 5, lane = tid & 31;
  const int wave_u = __builtin_amdgcn_readfirstlane(wave);  // wave-uniform
  const int mrow = lane & 15;
  const int koff = (lane < 16) ? 0 : 2;
  const float* A = (wave_u == 0) ? sA0 : &sA1[(wave_u - 1) * 16 * 64];
  const float* W = (wave_u == 0) ? sW0 : sW1;

  for (int nt = 0; nt < 4; ++nt) {
    const int ncol = nt * 16 + mrow;
    v8f c = {};
    for (int k0 = 0; k0 < 64; k0 += 4) {
      v2f a, b;
      a.x = A[mrow * 64 + k0 + koff];
      a.y = A[mrow * 64 + k0 + koff + 1];
      b.x = W[(k0 + koff) * 64 + ncol];
      b.y = W[(k0 + koff + 1) * 64 + ncol];
      c = wmma4(a, b, c);
    }
    const int mbase = (lane < 16) ? 0 : 8;
    if (wave_u == 0) {  // uniform branch: one s_cbranch, no per-store exec ops
#pragma unroll
      for (int r = 0; r < 8; ++r)
        x0[(size_t)(nbase + mbase + r) * 64 + ncol] = c[r] * 0.125f;
    } else {
#pragma unroll
      for (int r = 0; r < 8; ++r)
        x1[(size_t)(nbase + mbase + r) * 192 + ncol * 3 + (wave_u - 1)] =
            c[r] * 0.125f;
    }
  }
}

// ===========================================================================
// Kernel 2: edge path. Block: 256 thr (8 waves), each wave = 16 edges.
// Wr1/Wr2 async-staged RAW into LDS; the 1/sqrt8 * 1/sqrt32 scale is folded
// into the per-wave A fragments (4 VALU muls), 1/sqrt8 into t pre-softplus.
// Each 16x16 WMMA C tile of w is consumed immediately by quadrant
// (m00/m01/m10/m11) with gathers + atomicAdd into L2-resident agg buffers.
// ===========================================================================
__global__ __launch_bounds__(256) void k_edge(
    const float* __restrict__ emb, const float* __restrict__ eattr,
    const int* __restrict__ esrc, const int* __restrict__ edst,
    const float* __restrict__ Wr1, const float* __restrict__ Wr2,
    const float* __restrict__ x0, const float* __restrict__ x1,
    float* __restrict__ agg0, float* __restrict__ agg1) {
  __shared__ float sWr1[64];       // raw
  __shared__ float sWr2[8 * 256];  // raw
  __shared__ float sH[8][16 * 8];
  __shared__ int sSrc[8][16];
  __shared__ int sDst[8][16];
  __shared__ float sY[8][16 * 4];

  const int tid = threadIdx.x;
  // async-stage Wr2 (512 x b128) and Wr1 (16 x b128)
  {
    const unsigned l2 = lds_off(sWr2);
    const char* g2 = (const char*)Wr2;
    for (int i = tid; i < 512; i += 256) async_copy_b128(l2 + i * 16, g2 + i * 16);
    if (tid < 16)
      async_copy_b128(lds_off(sWr1) + tid * 16, (const char*)Wr1 + tid * 16);
  }
  wait_async0();
  __syncthreads();

  const int wave = tid >> 5, lane = tid & 31;
  const int ebase = blockIdx.x * 128 + wave * 16;

  if (lane < 16) {
    const int e = ebase + lane;
    const float4 e0 = ((const float4*)emb)[(size_t)e * 2 + 0];
    const float4 e1 = ((const float4*)emb)[(size_t)e * 2 + 1];
    float em[8] = {e0.x, e0.y, e0.z, e0.w, e1.x, e1.y, e1.z, e1.w};
    sSrc[wave][lane] = esrc[e];
    sDst[wave][lane] = edst[e];
    const float4 ya = ((const float4*)eattr)[e];
    sY[wave][lane * 4 + 0] = ya.x;
    sY[wave][lane * 4 + 1] = ya.y;
    sY[wave][lane * 4 + 2] = ya.z;
    sY[wave][lane * 4 + 3] = ya.w;
#pragma unroll
    for (int j = 0; j < 8; ++j) {
      float t = 0.f;
#pragma unroll
      for (int k = 0; k < 8; ++k) t += em[k] * sWr1[k * 8 + j];
      sH[wave][lane * 8 + j] = sspf(t * INV_SQRT8);
    }
  }
  __syncthreads();

  const int mrow = lane & 15;
  const int koff = (lane < 16) ? 0 : 2;
  const float WS = INV_SQRT8 * INV_SQRT32;  // w-scale + agg-scale, folded in A
  v2f aA, aB;
  aA.x = sH[wave][mrow * 8 + koff] * WS;
  aA.y = sH[wave][mrow * 8 + koff + 1] * WS;
  aB.x = sH[wave][mrow * 8 + 4 + koff] * WS;
  aB.y = sH[wave][mrow * 8 + 4 + koff + 1] * WS;

  for (int nt = 0; nt < 16; ++nt) {
    const int ncol = nt * 16 + mrow;
    v8f c = {};
    v2f b;
    b.x = sWr2[(0 + koff) * 256 + ncol];
    b.y = sWr2[(1 + koff) * 256 + ncol];
    c = wmma4(aA, b, c);
    b.x = sWr2[(4 + koff) * 256 + ncol];
    b.y = sWr2[(5 + koff) * 256 + ncol];
    c = wmma4(aB, b, c);

    const int q = nt >> 2;               // 0=m00 1=m01 2=m10 3=m11
    const int u = (nt & 3) * 16 + mrow;  // channel 0..63
#pragma unroll
    for (int r = 0; r < 8; ++r) {
      const int m = r + ((lane < 16) ? 0 : 8);
      const float wv = c[r];
      const int src = sSrc[wave][m];
      const int dst = sDst[wave][m];
      const float* y = &sY[wave][m * 4];
      if (q == 0) {  // m00 -> agg0[:, u]
        float val = wv * x0[(size_t)src * 64 + u] * y[0];
        atomicAdd(&agg0[(size_t)dst * 128 + u], val);
      } else if (q == 1) {  // m01 -> agg1[:, u, c]
        float base = wv * x0[(size_t)src * 64 + u];
        atomicAdd(&agg1[(size_t)dst * 384 + u * 3 + 0], base * y[1]);
        atomicAdd(&agg1[(size_t)dst * 384 + u * 3 + 1], base * y[2]);
        atomicAdd(&agg1[(size_t)dst * 384 + u * 3 + 2], base * y[3]);
      } else if (q == 2) {  // m10 -> agg1[:, 64+u, c]
        float wy = wv * y[0];
        const float* xr = &x1[(size_t)src * 192 + u * 3];
        atomicAdd(&agg1[(size_t)dst * 384 + (64 + u) * 3 + 0], wy * xr[0]);
        atomicAdd(&agg1[(size_t)dst * 384 + (64 + u) * 3 + 1], wy * xr[1]);
        atomicAdd(&agg1[(size_t)dst * 384 + (64 + u) * 3 + 2], wy * xr[2]);
      } else {  // m11 -> agg0[:, 64+u]
        const float* xr = &x1[(size_t)src * 192 + u * 3];
        float dot = xr[0] * y[1] + xr[1] * y[2] + xr[2] * y[3];
        atomicAdd(&agg0[(size_t)dst * 128 + 64 + u], wv * dot * INV_SQRT3);
      }
    }
  }
}

// ===========================================================================
// Kernel 3: out0/out1/sc0/sc1 + gating. Block: 128 thr (4 waves), 16 nodes.
// Wave q owns u-columns [16q,16q+16): 5 C tiles (S=scalars, G=gates,
// V0..V2=vector channels) across K=128 (agg@W2) and K=256 ((f (x) attrs)@Wsc);
// gating is element-local in registers. Weights read direct from L2.
// ===========================================================================
__global__ __launch_bounds__(128) void k_out(
    const float* __restrict__ f0, const float* __restrict__ f1,
    const float* __restrict__ attrs, const float* __restrict__ agg0,
    const float* __restrict__ agg1, const float* __restrict__ W20,
    const float* __restrict__ W21, const float* __restrict__ Wsc0,
    const float* __restrict__ Wsc1, float* __restrict__ out) {
  const int tid = threadIdx.x;
  const int wave = tid >> 5, lane = tid & 31;
  const int nbase = blockIdx.x * 16;
  const int mrow = lane & 15;
  const int koff = (lane < 16) ? 0 : 2;
  const int col = wave * 16 + mrow;  // output u-column, 0..63

  const float* a0row = agg0 + (size_t)(nbase + mrow) * 128;
  const float* a1row = agg1 + (size_t)(nbase + mrow) * 384;
  const float* f0row = f0 + (size_t)(nbase + mrow) * 64;
  const float* f1row = f1 + (size_t)(nbase + mrow) * 192;
  float at0 = attrs[(size_t)(nbase + mrow) * 4 + koff];
  float at1 = attrs[(size_t)(nbase + mrow) * 4 + koff + 1];

  v8f S = {}, G = {}, V0 = {}, V1 = {}, V2 = {};

  // ---- K=128: agg0@W2_0 (S,G) and agg1_c@W2_1 (V_c), scale 1/sqrt(128) ----
  for (int k0 = 0; k0 < 128; k0 += 4) {
    const int k = k0 + koff;
    v2f a, b;
    a.x = a0row[k] * INV_SQRT128;
    a.y = a0row[k + 1] * INV_SQRT128;
    b.x = W20[k * 128 + col];
    b.y = W20[(k + 1) * 128 + col];
    S = wmma4(a, b, S);
    b.x = W20[k * 128 + 64 + col];
    b.y = W20[(k + 1) * 128 + 64 + col];
    G = wmma4(a, b, G);
    v2f b2;
    b2.x = W21[k * 64 + col];
    b2.y = W21[(k + 1) * 64 + col];
    v2f a1;
    a1.x = a1row[k * 3 + 0] * INV_SQRT128;
    a1.y = a1row[(k + 1) * 3 + 0] * INV_SQRT128;
    V0 = wmma4(a1, b2, V0);
    a1.x = a1row[k * 3 + 1] * INV_SQRT128;
    a1.y = a1row[(k + 1) * 3 + 1] * INV_SQRT128;
    V1 = wmma4(a1, b2, V1);
    a1.x = a1row[k * 3 + 2] * INV_SQRT128;
    a1.y = a1row[(k + 1) * 3 + 2] * INV_SQRT128;
    V2 = wmma4(a1, b2, V2);
  }

  // ---- K=256: sc = (f (x) attrs) @ Wsc, k=(u*4+v), scale 1/sqrt(256) ----
  for (int k0 = 0; k0 < 256; k0 += 4) {
    const int u = k0 >> 2;
    const int k = k0 + koff;  // v = koff, koff+1 -> at0/at1
    float fv = f0row[u] * INV_SQRT256;
    v2f a, b;
    a.x = fv * at0;
    a.y = fv * at1;
    b.x = Wsc0[k * 128 + col];
    b.y = Wsc0[(k + 1) * 128 + col];
    S = wmma4(a, b, S);
    b.x = Wsc0[k * 128 + 64 + col];
    b.y = Wsc0[(k + 1) * 128 + 64 + col];
    G = wmma4(a, b, G);
    v2f b2;
    b2.x = Wsc1[k * 64 + col];
    b2.y = Wsc1[(k + 1) * 64 + col];
    float f1v = f1row[u * 3 + 0] * INV_SQRT256;
    a.x = f1v * at0;
    a.y = f1v * at1;
    V0 = wmma4(a, b2, V0);
    f1v = f1row[u * 3 + 1] * INV_SQRT256;
    a.x = f1v * at0;
    a.y = f1v * at1;
    V1 = wmma4(a, b2, V1);
    f1v = f1row[u * 3 + 2] * INV_SQRT256;
    a.x = f1v * at0;
    a.y = f1v * at1;
    V2 = wmma4(a, b2, V2);
  }

  // ---- gating + residual + concat store ----
#pragma unroll
  for (int r = 0; r < 8; ++r) {
    const int m = r + ((lane < 16) ? 0 : 8);
    const int n = nbase + m;
    const int u = col;
    float gate = sspf(G[r]);
    out[(size_t)n * 256 + u] = f0[(size_t)n * 64 + u] + sspf(S[r]);
    const float* fr = &f1[(size_t)n * 192 + u * 3];
    float* orow = &out[(size_t)n * 256 + 64 + u * 3];
    orow[0] = fr[0] + V0[r] * gate;
    orow[1] = fr[1] + V1[r] * gate;
    orow[2] = fr[2] + V2[r] * gate;
  }
}

// ===========================================================================
extern "C" void kernel_launch(void* const* d_in, const int* in_sizes, int n_in,
                              void* d_out, int out_size, void* d_ws,
                              size_t ws_size, hipStream_t stream) {
  const float* f0 = (const float*)d_in[0];
  const float* f1 = (const float*)d_in[1];
  const float* attrs = (const float*)d_in[2];
  const float* emb = (const float*)d_in[3];
  const float* eatt = (const float*)d_in[4];
  const int* esrc = (const int*)d_in[5];
  const int* edst = (const int*)d_in[6];
  const float* W10 = (const float*)d_in[7];
  const float* W11 = (const float*)d_in[8];
  const float* Wr1 = (const float*)d_in[9];
  const float* Wr2 = (const float*)d_in[10];
  const float* W20 = (const float*)d_in[11];
  const float* W21 = (const float*)d_in[12];
  const float* Wsc0 = (const float*)d_in[13];
  const float* Wsc1 = (const float*)d_in[14];

  float* ws = (float*)d_ws;
  float* x0 = ws;              //   640,000 f
  float* x1 = ws + 640000;     // 1,920,000 f
  float* agg0 = ws + 2560000;  // 1,280,000 f (zeroed each launch)
  float* agg1 = ws + 3840000;  // 3,840,000 f (zeroed each launch)

  hipMemsetAsync(agg0, 0, (size_t)5120000 * sizeof(float), stream);

  k_node_linear<<<NN / 16, 128, 0, stream>>>(f0, f1, W10, W11, x0, x1);
  k_edge<<<NE / 128, 256, 0, stream>>>(emb, eatt, esrc, edst, Wr1, Wr2, x0, x1,
                                       agg0, agg1);
  k_out<<<NN / 16, 128, 0, stream>>>(f0, f1, attrs, agg0, agg1, W20, W21, Wsc0,
                                     Wsc1, (float*)d_out);
}